// OptionAttentionSum_7919919694186
// MI455X (gfx1250) — compile-verified
//
#include <hip/hip_runtime.h>
#include <hip/hip_bf16.h>
#include <stdint.h>

// Problem constants from the reference: B=128, L=4096, O=10, W=5.
#define OA_L  4096
#define OA_O  10
#define OA_W  5
#define OA_OW (OA_O * OA_W)
#define OA_THREADS 256

// One workgroup per batch row.
// Stage doc indices+probs into LDS via CDNA5 async global->LDS DMA (ASYNCcnt),
// then all 50 option-words re-read the staged tile (50x reuse, HBM read-once).
__global__ __launch_bounds__(OA_THREADS) void
OptionAttentionSum_kernel(const int* __restrict__ doc_idx,
                          const float* __restrict__ doc_prob,
                          const int* __restrict__ options,
                          float* __restrict__ out)
{
    __shared__ __align__(16) int   sIdx[OA_L];
    __shared__ __align__(16) float sProb[OA_L];
    __shared__ float sWord[OA_OW];

    const int b    = blockIdx.x;
    const int tid  = threadIdx.x;          // 0..255
    const int lane = tid & 31;             // wave32
    const int wave = tid >> 5;             // 0..7

    // ---- Stage 16KB idx + 16KB prob into LDS with async b128 loads ----
    {
        // Low 32 bits of a generic pointer to LDS == byte offset in the
        // workgroup's LDS allocation (CDNA5 LDS aperture truncation).
        const unsigned lbaseI = (unsigned)(uintptr_t)(void*)sIdx;
        const unsigned lbaseP = (unsigned)(uintptr_t)(void*)sProb;
        const unsigned long long gI =
            (unsigned long long)(uintptr_t)(doc_idx + (size_t)b * OA_L);
        const unsigned long long gP =
            (unsigned long long)(uintptr_t)(doc_prob + (size_t)b * OA_L);

        // OA_L*4 bytes = 16384; 256 lanes * 16B = 4096B per sweep -> 4 sweeps.
        // s_wait_xcnt 0x0 inside each asm block: the address VGPRs of an
        // in-flight VMEM op may be re-read on XNACK replay (tracked by XCNT);
        // clang can't see that for inline asm, so drain XCNT before letting
        // the register allocator recycle the operands.
        #pragma unroll
        for (int s = 0; s < (OA_L * 4) / (OA_THREADS * 16); ++s) {
            const unsigned byteOff = (unsigned)(s * (OA_THREADS * 16) + tid * 16);
            const unsigned           ldsI = lbaseI + byteOff;
            const unsigned long long gaI  = gI + byteOff;
            asm volatile("global_load_async_to_lds_b128 %0, %1, off\n\t"
                         "s_wait_xcnt 0x0"
                         :: "v"(ldsI), "v"(gaI) : "memory");
            const unsigned           ldsP = lbaseP + byteOff;
            const unsigned long long gaP  = gP + byteOff;
            asm volatile("global_load_async_to_lds_b128 %0, %1, off\n\t"
                         "s_wait_xcnt 0x0"
                         :: "v"(ldsP), "v"(gaP) : "memory");
        }
        // Each wave waits for its own async transfers, then the workgroup
        // barrier makes the full tile visible to all waves.
        asm volatile("s_wait_asynccnt 0x0" ::: "memory");
    }
    __syncthreads();

    // ---- Per option-word match-sum over the staged tile ----
    const int* opt_b = options + (size_t)b * OA_OW;
    const int4*   vIdx  = (const int4*)sIdx;
    const float4* vProb = (const float4*)sProb;

    for (int ow = wave; ow < OA_OW; ow += (OA_THREADS / 32)) {
        const int wv = opt_b[ow];          // wave-uniform scalar
        float acc = 0.0f;
        #pragma unroll 4
        for (int i = lane; i < OA_L / 4; i += 32) {
            const int4   ii = vIdx[i];     // ds_load_b128, conflict-free
            const float4 pp = vProb[i];
            acc += (ii.x == wv) ? pp.x : 0.0f;
            acc += (ii.y == wv) ? pp.y : 0.0f;
            acc += (ii.z == wv) ? pp.z : 0.0f;
            acc += (ii.w == wv) ? pp.w : 0.0f;
        }
        // wave32 butterfly reduction
        #pragma unroll
        for (int off = 16; off > 0; off >>= 1)
            acc += __shfl_xor(acc, off, 32);
        if (lane == 0)
            sWord[ow] = acc;
    }
    __syncthreads();

    // ---- Combine words per option, divide by #non-padding words ----
    if (tid < OA_O) {
        float s = 0.0f, cnt = 0.0f;
        #pragma unroll
        for (int w = 0; w < OA_W; ++w) {
            s   += sWord[tid * OA_W + w];
            cnt += (opt_b[tid * OA_W + w] != 0) ? 1.0f : 0.0f;
        }
        out[(size_t)b * OA_O + tid] = s / cnt;   // x/0 -> inf/nan like reference
    }
}

extern "C" void kernel_launch(void* const* d_in, const int* in_sizes, int n_in,
                              void* d_out, int out_size, void* d_ws, size_t ws_size,
                              hipStream_t stream) {
    (void)n_in; (void)d_ws; (void)ws_size;
    const int*   doc_idx  = (const int*)d_in[0];   // (B, L) int32 (JAX-canonicalized)
    const float* doc_prob = (const float*)d_in[1]; // (B, L) f32
    const int*   options  = (const int*)d_in[2];   // (B, O, W) int32
    float*       out      = (float*)d_out;         // (B, O) f32

    const int B = out_size / OA_O;                 // 128
    (void)in_sizes;

    hipLaunchKernelGGL(OptionAttentionSum_kernel,
                       dim3(B), dim3(OA_THREADS), 0, stream,
                       doc_idx, doc_prob, options, out);
}